// Triplet_Sampling_53223234732451
// MI455X (gfx1250) — compile-verified
//
#include <hip/hip_runtime.h>
#include <hip/hip_bf16.h>

// Triplet semi-hard mining, fused. N=8192, D=256, TRIALS=150.
// B-side candidate rows are gathered by the Tensor Data Mover (gather mode,
// 16x 16-bit row indices) into LDS, double-buffered per wave; the 16x16 dot
// tiles run on v_wmma_f32_16x16x4_f32 fed by ds_load_b64 (exact f32 math).

#define N_BATCH 8192
#define DIM     256
#define TRIALS  150
#define SLOTS   151            // 150 negative trials + 1 positive (t==150)
#define SSTRIDE 152            // score row stride (pad)

typedef __attribute__((ext_vector_type(2))) float    v2f;
typedef __attribute__((ext_vector_type(8))) float    v8f;
typedef __attribute__((ext_vector_type(4))) unsigned v4u;
typedef __attribute__((ext_vector_type(8))) int      v8i;
typedef __attribute__((ext_vector_type(4))) int      v4i;

// ---- dynamic LDS layout (bytes) ----
#define SA_OFF    0                       // A tile: 16 rows x 256 f32 = 16 KB
#define SB_OFF    16384                   // B bufs: 8 waves x 2 x 4 KB = 64 KB
#define VBUF_OFF  (16384 + 65536)         // 16*152 f32
#define CBUF_OFF  (VBUF_OFF + 16 * SSTRIDE * 4)
#define SEL_OFF   (CBUF_OFF + 16 * SSTRIDE * 4)
#define SMEM_BYTES (SEL_OFF + 128)

// ---------------- kernel 1: n_genu = count(tgt == 0) ----------------
__global__ void ts_count_genuine(const int* __restrict__ tgt,
                                 int* __restrict__ n_genu_out) {
    __shared__ int red[256];
    int s = 0;
    for (int i = threadIdx.x; i < N_BATCH; i += 256) s += (tgt[i] == 0) ? 1 : 0;
    red[threadIdx.x] = s;
    __syncthreads();
    for (int st = 128; st > 0; st >>= 1) {
        if ((int)threadIdx.x < st) red[threadIdx.x] += red[threadIdx.x + st];
        __syncthreads();
    }
    if (threadIdx.x == 0) n_genu_out[0] = red[0];
}

// ---------------- kernel 2: sq[r] = ||hd[r]||^2 ----------------
__global__ void ts_row_sq(const float* __restrict__ hd, float* __restrict__ sq) {
    const int lane = threadIdx.x & 31;
    const int wave = threadIdx.x >> 5;
    const int row  = blockIdx.x * 8 + wave;
    const float* p = hd + (size_t)row * DIM + lane * 8;
    float4 x = *(const float4*)p;
    float4 y = *(const float4*)(p + 4);
    float s = x.x * x.x + x.y * x.y + x.z * x.z + x.w * x.w
            + y.x * y.x + y.y * y.y + y.z * y.z + y.w * y.w;
    #pragma unroll
    for (int m = 16; m >= 1; m >>= 1) s += __shfl_xor(s, m, 32);
    if (lane == 0) sq[row] = s;
}

// Candidate column for (anchor i, slot t). t==150 => positive pick.
__device__ __forceinline__ int ts_cand(int t, int i, int len_neg, int len_pos,
                                       const float* __restrict__ u_neg,
                                       const float* __restrict__ u_pos) {
    if (t < TRIALS) {
        float u = u_neg[(size_t)i * TRIALS + t];
        int c0 = (int)(u * (float)len_neg);
        return (c0 < len_neg - 1) ? c0 : (len_neg - 1);
    } else {
        float u = u_pos[i];
        int excl = (i < len_pos) ? 1 : 0;
        int eff  = len_pos - excl;
        int c0 = (int)(u * (float)eff);
        c0 = (c0 < eff - 1) ? c0 : (eff - 1);
        return c0 + ((excl && (c0 >= i)) ? 1 : 0);
    }
}

// TDM gather: 16 rows (16-bit indices in g2/g3), each 64 f32, from a row-major
// [8192 x 256] f32 tensor starting at gbase (column offset pre-applied), into
// LDS at byte offset lds_off (rows stored compacted, 256 B apart).
__device__ __forceinline__ void tdm_gather16(unsigned lds_off,
                                             const float* gbase,
                                             v4i g2, v4i g3) {
    unsigned long long ga = (unsigned long long)(uintptr_t)gbase;
    v4u g0 = { 0x80000001u,                                  // count=1, gather, idx16
               lds_off,
               (unsigned)(ga & 0xFFFFFFFFu),
               (unsigned)((ga >> 32) & 0x01FFFFFFu) | 0x80000000u };  // type=2
    v8i g1 = { 0x00020000,   // data_size = 4 B
               0x00400000,   // tensor_dim0 = 64 (low 16 in [31:16])
               0x20000000,   // tensor_dim1 = 8192 (low 16 in [31:16])
               0x00400000,   // tile_dim0 = 64 in [31:16]
               16,           // tile_dim1 = 16 valid indices
               256,          // tensor_dim0_stride = 256 elements
               0, 0 };
    v8i g4 = { 0, 0, 0, 0, 0, 0, 0, 0 };                     // VADDR4 unused
    __builtin_amdgcn_tensor_load_to_lds(g0, g1, g2, g3, g4, 0);
}

// ---------------- kernel 3: fused mining + output gather ----------------
// Block = 16 anchors, 256 threads = 8 waves. Wave w handles slots t = w, w+8, ...
__global__ void ts_mine(const float* __restrict__ hd,
                        const int*   __restrict__ tgt,
                        const float* __restrict__ margin_p,
                        const float* __restrict__ u_pos,
                        const float* __restrict__ u_neg,
                        const int*   __restrict__ n_genu_p,
                        const float* __restrict__ sq,
                        float* __restrict__ out) {
    extern __shared__ char smem[];
    float* sA   = (float*)(smem + SA_OFF);        // 16 x 256 anchor tile
    float* vbuf = (float*)(smem + VBUF_OFF);
    int*   cbuf = (int*)  (smem + CBUF_OFF);
    int*   sel_pos = (int*)(smem + SEL_OFF);
    int*   sel_neg = (int*)(smem + SEL_OFF + 64);

    const int tid  = threadIdx.x;
    const int lane = tid & 31;
    const int wave = tid >> 5;
    const int i0   = blockIdx.x * 16;

    // ---- stage A tile (16 anchor rows, contiguous 16 KB) ----
    {
        const float4* src = (const float4*)(hd + (size_t)i0 * DIM);
        float4* dst = (float4*)sA;
        for (int q = tid; q < 16 * DIM / 4; q += 256) dst[q] = src[q];
    }
    __syncthreads();

    // Per-lane fixed role: tile column n == anchor (i0+n).
    const int n    = lane & 15;
    const int i    = i0 + n;
    const int kofs = (lane < 16) ? 0 : 2;          // f32 16x16x4 A/B lane->K split

    const int ngen   = n_genu_p[0];
    const int nspoof = N_BATCH - ngen;
    const int tgt_i  = tgt[i];
    const int len_neg = (tgt_i == 0) ? nspoof : ngen;
    const int len_pos = (tgt_i == 0) ? ngen   : nspoof;

    // Per-wave double buffers (byte offsets in LDS)
    const unsigned bboff0 = SB_OFF + wave * 8192;
    const unsigned bboff1 = bboff0 + 4096;
    const float* sB0 = (const float*)(smem + bboff0);
    const float* sB1 = (const float*)(smem + bboff1);

    for (int t = wave; t < SLOTS; t += 8) {
        // ---- candidate index, packed into 16-bit gather indices ----
        const int j = ts_cand(t, i, len_neg, len_pos, u_neg, u_pos);
        unsigned p[8];
        #pragma unroll
        for (int q = 0; q < 8; ++q) {
            unsigned lo = (unsigned)__builtin_amdgcn_readlane(j, 2 * q)     & 0xFFFFu;
            unsigned hi = (unsigned)__builtin_amdgcn_readlane(j, 2 * q + 1) & 0xFFFFu;
            p[q] = lo | (hi << 16);
        }
        const v4i g2 = { (int)p[0], (int)p[1], (int)p[2], (int)p[3] };
        const v4i g3 = { (int)p[4], (int)p[5], (int)p[6], (int)p[7] };

        // ---- TDM double-buffered pipeline over 4 K-chunks of 64 ----
        tdm_gather16(bboff0, hd + 0 * 64, g2, g3);
        v8f acc = {0.f, 0.f, 0.f, 0.f, 0.f, 0.f, 0.f, 0.f};
        #pragma unroll
        for (int c = 0; c < 4; ++c) {
            if (c < 3) {
                tdm_gather16((c & 1) ? bboff0 : bboff1, hd + (c + 1) * 64, g2, g3);
                __builtin_amdgcn_s_wait_tensorcnt(1);   // chunk c complete
            } else {
                __builtin_amdgcn_s_wait_tensorcnt(0);   // last chunk complete
            }
            const float* sB = (c & 1) ? sB1 : sB0;
            #pragma unroll
            for (int s = 0; s < 16; ++s) {
                v2f a = *(const v2f*)(sA + n * DIM + c * 64 + s * 4 + kofs);
                v2f b = *(const v2f*)(sB + n * 64  +           s * 4 + kofs);
                acc = __builtin_amdgcn_wmma_f32_16x16x4_f32(
                    false, a, false, b, (short)0, acc, false, false);
            }
        }

        // ---- extract diagonal D[c][c] ----
        // c<8 : VGPR c, lane c (lanes 0..7); c>=8 : VGPR c-8, lane c+16 (24..31)
        int r = (lane < 16) ? lane : (lane - 24);
        float val = acc[0];
        val = (r == 1) ? acc[1] : val;
        val = (r == 2) ? acc[2] : val;
        val = (r == 3) ? acc[3] : val;
        val = (r == 4) ? acc[4] : val;
        val = (r == 5) ? acc[5] : val;
        val = (r == 6) ? acc[6] : val;
        val = (r == 7) ? acc[7] : val;

        if (lane < 8 || lane >= 24) {
            vbuf[n * SSTRIDE + t] = sq[j] - 2.0f * val;   // dst minus sq[i] (const)
            cbuf[n * SSTRIDE + t] = j;
        }
    }
    __syncthreads();

    // ---- per-anchor sequential selection (reference break semantics) ----
    if (tid < 16) {
        const float marg = margin_p[0];
        const float vpos = vbuf[tid * SSTRIDE + TRIALS];
        int hit = -1, besti = 0;
        float best = vbuf[tid * SSTRIDE + 0];
        for (int tt = 0; tt < TRIALS; ++tt) {
            float x = vbuf[tid * SSTRIDE + tt];
            if (hit < 0 && (vpos < x) && ((vpos - x + marg) > 0.f)) hit = tt;
            if (x < best) { best = x; besti = tt; }   // strict '<' == first argmin
        }
        int tsel = (hit >= 0) ? hit : besti;
        sel_pos[tid] = cbuf[tid * SSTRIDE + TRIALS];
        sel_neg[tid] = cbuf[tid * SSTRIDE + tsel];
    }
    __syncthreads();

    // ---- gather output [16 anchors][3][256] as float4 (12 iters/thread) ----
    for (int idx = tid; idx < 16 * 3 * (DIM / 4); idx += 256) {
        int c    = idx / (3 * (DIM / 4));
        int rem  = idx % (3 * (DIM / 4));
        int part = rem / (DIM / 4);
        int off4 = rem % (DIM / 4);
        int srow = (part == 0) ? (i0 + c) : ((part == 1) ? sel_pos[c] : sel_neg[c]);
        float4 v = *(const float4*)(hd + (size_t)srow * DIM + off4 * 4);
        *(float4*)(out + ((size_t)(i0 + c) * 3 + part) * DIM + off4 * 4) = v;
    }
}

extern "C" void kernel_launch(void* const* d_in, const int* in_sizes, int n_in,
                              void* d_out, int out_size, void* d_ws, size_t ws_size,
                              hipStream_t stream) {
    const float* hd     = (const float*)d_in[0];   // [8192,256] f32
    const int*   tgt    = (const int*)  d_in[1];   // [8192] i32
    const float* margin = (const float*)d_in[2];   // [1] f32
    const float* u_pos  = (const float*)d_in[3];   // [8192] f32
    const float* u_neg  = (const float*)d_in[4];   // [8192,150] f32
    float*       out    = (float*)d_out;           // [8192,3,256] f32

    int*   ws_ng = (int*)d_ws;                     // ws[0]: n_genu
    float* ws_sq = (float*)d_ws + 8;               // ws[8..8+8191]: row norms

    ts_count_genuine<<<1, 256, 0, stream>>>(tgt, ws_ng);
    ts_row_sq<<<N_BATCH / 8, 256, 0, stream>>>(hd, ws_sq);
    ts_mine<<<N_BATCH / 16, 256, SMEM_BYTES, stream>>>(hd, tgt, margin, u_pos,
                                                       u_neg, ws_ng, ws_sq, out);
}